// Core_3324304687622
// MI455X (gfx1250) — compile-verified
//
#include <hip/hip_runtime.h>

// ---------------------------------------------------------------------------
// Sizes (match reference constants)
// ---------------------------------------------------------------------------
#define MEME_DIM        64
#define NUM_NODES       8192
#define VISION_MEMES    2048
#define INPUTS_PER_NODE 32
#define VOCAB_SIZE      256
#define TOTAL_G         (NUM_NODES * MEME_DIM)   // 524288

typedef float v2f __attribute__((ext_vector_type(2)));
typedef float v8f __attribute__((ext_vector_type(8)));

// ---------------------------------------------------------------------------
// Kernel 1: gather + mean over K, per-node partial sums for global stats
// one block per node, 64 threads (one per meme dim)
// ---------------------------------------------------------------------------
__global__ __launch_bounds__(64) void k_gather(
    const float* __restrict__ x,        // [2048, 64]
    const float* __restrict__ memes,    // [8192, 64]
    const int*   __restrict__ idx,      // [8192, 32]
    float* __restrict__ g_raw,          // [8192, 64]
    float* __restrict__ ps,             // [8192] partial sum
    float* __restrict__ ps2)            // [8192] partial sumsq
{
    __shared__ float red[64];
    const int n = blockIdx.x;
    const int e = threadIdx.x;
    const int* ip = idx + n * INPUTS_PER_NODE;

    float acc = 0.0f;
    #pragma unroll
    for (int k = 0; k < INPUTS_PER_NODE; ++k) {
        int r = ip[k];
        const float* src = (r < VISION_MEMES)
                         ? (x + (size_t)r * MEME_DIM)
                         : (memes + (size_t)(r - VISION_MEMES) * MEME_DIM);
        acc += src[e];
    }
    float g = acc * (1.0f / 32.0f);
    g_raw[(size_t)n * MEME_DIM + e] = g;

    // deterministic tree reductions (no float atomics -> replay-stable)
    red[e] = g; __syncthreads();
    for (int s = 32; s > 0; s >>= 1) { if (e < s) red[e] += red[e + s]; __syncthreads(); }
    if (e == 0) ps[n] = red[0];
    __syncthreads();
    red[e] = g * g; __syncthreads();
    for (int s = 32; s > 0; s >>= 1) { if (e < s) red[e] += red[e + s]; __syncthreads(); }
    if (e == 0) ps2[n] = red[0];
}

// ---------------------------------------------------------------------------
// Kernel 2: reduce partials -> global mean and 1/std (ddof=1)
// ---------------------------------------------------------------------------
__global__ __launch_bounds__(256) void k_stats(
    const float* __restrict__ ps,
    const float* __restrict__ ps2,
    float* __restrict__ scal)           // [0]=mean, [1]=1/std
{
    __shared__ float r1[256];
    __shared__ float r2[256];
    const int t = threadIdx.x;
    float a = 0.0f, b = 0.0f;
    for (int i = t; i < NUM_NODES; i += 256) { a += ps[i]; b += ps2[i]; }
    r1[t] = a; r2[t] = b; __syncthreads();
    for (int s = 128; s > 0; s >>= 1) {
        if (t < s) { r1[t] += r1[t + s]; r2[t] += r2[t + s]; }
        __syncthreads();
    }
    if (t == 0) {
        const float cnt  = (float)TOTAL_G;
        float mean = r1[0] / cnt;
        float var  = (r2[0] - cnt * mean * mean) / (cnt - 1.0f);
        scal[0] = mean;
        scal[1] = 1.0f / sqrtf(var);    // reference divides by std with no eps
    }
}

// ---------------------------------------------------------------------------
// Kernel 3: fused per-node remix. One block (128 thr = 4 waves) per node.
//   einsum1: dot[v] = sum_e g[e]*vocab[e,v]      (V_WMMA_F32_16X16X4_F32 chains)
//   fractions: standardize over v (ddof=1), tanh soft-clip, pow-sharpen, norm
//   einsum2: out[e] = sum_v vocab[e,v]*frac[v]   (WMMA chains; vocab re-read
//            hits WGP$/L2 -> HBM sees vocab ~once)
// ---------------------------------------------------------------------------
__device__ __forceinline__ float block_reduce128(float v, float* red, int tid) {
    red[tid] = v; __syncthreads();
    for (int s = 64; s > 0; s >>= 1) { if (tid < s) red[tid] += red[tid + s]; __syncthreads(); }
    float r = red[0]; __syncthreads();
    return r;
}

__global__ __launch_bounds__(128) void k_node(
    const float* __restrict__ g_raw,    // [8192, 64]
    const float* __restrict__ vocab,    // [8192, 64, 256]
    const float* __restrict__ scal,     // [mean, 1/std]
    const float* __restrict__ raw_sc,   // [1]
    const float* __restrict__ raw_sh,   // [1]
    float* __restrict__ out)            // [8192, 64]
{
    __shared__ float gn[MEME_DIM];      // standardized g row
    __shared__ float dv[VOCAB_SIZE];    // dot products -> fractions (in place)
    __shared__ float red[128];
    __shared__ float outst[MEME_DIM];

    const int n    = blockIdx.x;
    const int tid  = threadIdx.x;
    const int lane = tid & 31;
    const int wave = tid >> 5;
    const int mrow = lane & 15;          // M position within WMMA tile
    const int koff = (lane >> 4) << 1;   // K pair: 0 for lanes 0-15, 2 for 16-31
    const float* vp = vocab + (size_t)n * (MEME_DIM * VOCAB_SIZE);

    // standardized input row
    if (tid < MEME_DIM) {
        float mu = scal[0], rs = scal[1];
        gn[tid] = (g_raw[(size_t)n * MEME_DIM + tid] - mu) * rs;
    }
    __syncthreads();

    // ---- einsum1: dot[v], 16 columns per WMMA chain, 4 chains per wave ----
    for (int q = 0; q < 4; ++q) {
        const int vt   = wave * 4 + q;
        const int vcol = vt * 16 + mrow;
        v8f acc = {};
        #pragma unroll
        for (int eb = 0; eb < MEME_DIM; eb += 4) {
            v2f a, b;
            // A[m,k] = vocab[eb+k, vt*16+m]
            a.x = vp[(size_t)(eb + koff)     * VOCAB_SIZE + vcol];
            a.y = vp[(size_t)(eb + koff + 1) * VOCAB_SIZE + vcol];
            // B[k,n] = gn[eb+k] (broadcast over n)
            b.x = gn[eb + koff];
            b.y = gn[eb + koff + 1];
            acc = __builtin_amdgcn_wmma_f32_16x16x4_f32(
                false, a, false, b, (short)0, acc, false, false);
        }
        // D[m, n] = dot[vt*16+m] for any n; lane0 holds M=0..7, lane16 M=8..15
        #pragma unroll
        for (int r = 0; r < 8; ++r) {
            if (lane == 0)  dv[vt * 16 + r]     = acc[r];
            if (lane == 16) dv[vt * 16 + 8 + r] = acc[r];
        }
    }
    __syncthreads();

    // ---- fractions: per-node standardize (ddof=1) + soft-clip + sharpen ----
    const float sc  = __expf(raw_sc[0]);                  // soft_clip
    const float l2s = raw_sh[0] * 1.4426950408889634f;    // log2(sharpness) = raw*log2(e)

    float d0 = dv[tid];
    float d1 = dv[tid + 128];
    float s  = block_reduce128(d0 + d1, red, tid);
    float s2 = block_reduce128(d0 * d0 + d1 * d1, red, tid);
    float mean = s * (1.0f / (float)VOCAB_SIZE);
    float var  = (s2 - (float)VOCAB_SIZE * mean * mean) * (1.0f / (VOCAB_SIZE - 1.0f));
    float inv  = 1.0f / (sqrtf(fmaxf(var, 0.0f)) + 0.001f);
    d0 = (d0 - mean) * inv;
    d1 = (d1 - mean) * inv;
    d0 = tanhf(d0 / sc) * sc;
    d1 = tanhf(d1 / sc) * sc;
    float p0 = exp2f(d0 * l2s);
    float p1 = exp2f(d1 * l2s);
    float psum = block_reduce128(p0 + p1, red, tid);
    float dnm  = 1.0f / (psum + 0.001f);
    dv[tid]       = p0 * dnm;
    dv[tid + 128] = p1 * dnm;
    __syncthreads();

    // ---- einsum2: out[e] = sum_v vocab[e,v]*frac[v], one e-tile per wave ----
    {
        const int et  = wave;
        const int row = et * 16 + mrow;
        const float* rp = vp + (size_t)row * VOCAB_SIZE;
        v8f acc = {};
        #pragma unroll
        for (int vb = 0; vb < VOCAB_SIZE; vb += 4) {
            v2f a, b;
            // A[m,k] = vocab[et*16+m, vb+k]  (two consecutive floats per lane)
            a.x = rp[vb + koff];
            a.y = rp[vb + koff + 1];
            // B[k,n] = frac[vb+k]
            b.x = dv[vb + koff];
            b.y = dv[vb + koff + 1];
            acc = __builtin_amdgcn_wmma_f32_16x16x4_f32(
                false, a, false, b, (short)0, acc, false, false);
        }
        #pragma unroll
        for (int r = 0; r < 8; ++r) {
            if (lane == 0)  outst[et * 16 + r]     = acc[r];
            if (lane == 16) outst[et * 16 + 8 + r] = acc[r];
        }
    }
    __syncthreads();
    if (tid < MEME_DIM) out[(size_t)n * MEME_DIM + tid] = outst[tid];
}

// ---------------------------------------------------------------------------
// Launch
// ---------------------------------------------------------------------------
extern "C" void kernel_launch(void* const* d_in, const int* in_sizes, int n_in,
                              void* d_out, int out_size, void* d_ws, size_t ws_size,
                              hipStream_t stream) {
    const float* x      = (const float*)d_in[0];
    const float* memes  = (const float*)d_in[1];
    const int*   idx    = (const int*)  d_in[2];
    const float* vocab  = (const float*)d_in[3];
    const float* raw_sc = (const float*)d_in[4];
    const float* raw_sh = (const float*)d_in[5];
    float* outp = (float*)d_out;

    float* wsf   = (float*)d_ws;
    float* g_raw = wsf;                          // 524288 floats
    float* ps    = wsf + TOTAL_G;                // 8192
    float* ps2   = wsf + TOTAL_G + NUM_NODES;    // 8192
    float* scal  = wsf + TOTAL_G + 2 * NUM_NODES;// 2

    k_gather<<<NUM_NODES, 64, 0, stream>>>(x, memes, idx, g_raw, ps, ps2);
    k_stats <<<1, 256, 0, stream>>>(ps, ps2, scal);
    k_node  <<<NUM_NODES, 128, 0, stream>>>(g_raw, vocab, scal, raw_sc, raw_sh, outp);
}